// Decoder_63067299775239
// MI455X (gfx1250) — compile-verified
//
#include <hip/hip_runtime.h>
#include <hip/hip_bf16.h>

typedef __attribute__((ext_vector_type(2))) float v2f;
typedef __attribute__((ext_vector_type(8))) float v8f;

constexpr int kD = 128;  // embedding dim, fixed by the reference

// -----------------------------------------------------------------------------
// Phase 1: per-node scores via fp32 WMMA.
//   S = emb (N x 128) @ W^T (128 x 2)  -> s_tab[n], t_tab[n]
// Each wave handles a 16-node tile: C(16x16) accumulated over 32 steps of
// V_WMMA_F32_16X16X4_F32; B has w_src in column 0, w_dst in column 1.
// -----------------------------------------------------------------------------
__global__ __launch_bounds__(256) void node_score_wmma(
    const float* __restrict__ emb,     // N x 128
    const float* __restrict__ weight,  // 1 x 256 (w_src | w_dst)
    float* __restrict__ s_tab,         // N
    float* __restrict__ t_tab,         // N
    int N, int numTiles)
{
    const int wavesPerBlock = blockDim.x >> 5;
    const int tile = blockIdx.x * wavesPerBlock + (threadIdx.x >> 5);
    if (tile >= numTiles) return;  // wave-uniform: EXEC stays all-1s past here

    const int lane = threadIdx.x & 31;
    const int half = lane >> 4;   // 0: K = k0,k0+1 ; 1: K = k0+2,k0+3
    const int l    = lane & 15;   // A: row within tile ; B: column N
    const int tileBase = tile * 16;

    // A-matrix per-lane row pointer (clamped for a possible ragged tail tile).
    int row = tileBase + l;
    if (row >= N) row = N - 1;
    const float* rowPtr = emb + (size_t)row * kD + 2 * half;

    // B-matrix: column l of B is w_src (l==0) / w_dst (l==1) / zero (l>=2).
    // Branchless: always load from a safe column (l&1) and mask lanes l>=2,
    // so no divergence surrounds the WMMA ops.
    const float bmask = (l < 2) ? 1.0f : 0.0f;
    const float* wPtr = weight + (size_t)(l & 1) * kD + 2 * half;

    v8f c = {};
#pragma unroll
    for (int k0 = 0; k0 < kD; k0 += 4) {
        v2f a = *(const v2f*)(rowPtr + k0);   // (K=k0+2h, k0+2h+1) of row
        v2f bw = *(const v2f*)(wPtr + k0);
        v2f b;
        b.x = bw.x * bmask;
        b.y = bw.y * bmask;
        // D = A(16x4) * B(4x16) + C ; fp32 throughout
        c = __builtin_amdgcn_wmma_f32_16x16x4_f32(
            /*neg_a=*/false, a, /*neg_b=*/false, b,
            /*c_mod=*/(short)0, c, /*reuse_a=*/false, /*reuse_b=*/false);
    }

    // C layout (16x16 f32): VGPR v, lanes 0-15 -> (M=v, N=lane);
    // lanes 16-31 -> (M=v+8, N=lane-16).
    // Column 0 (= s) lives in lanes 0 and 16; column 1 (= t) in lanes 1 and 17.
    const int mBase = tileBase + half * 8;
    if (l == 0) {
#pragma unroll
        for (int v = 0; v < 8; ++v) {
            int m = mBase + v;
            if (m < N) s_tab[m] = c[v];
        }
    } else if (l == 1) {
#pragma unroll
        for (int v = 0; v < 8; ++v) {
            int m = mBase + v;
            if (m < N) t_tab[m] = c[v];
        }
    }
}

// -----------------------------------------------------------------------------
// Phase 2: per-edge logit = s[src] + t[dst] + bias.
// Streaming b128 loads of (src,dst) int64 pairs; s/t tables (800 KB total)
// stay L2-resident, so the random gathers are cheap.
// -----------------------------------------------------------------------------
struct alignas(16) EdgePair { long long src, dst; };

__global__ __launch_bounds__(256) void edge_logits_kernel(
    const EdgePair* __restrict__ edges,
    const float* __restrict__ s_tab,
    const float* __restrict__ t_tab,
    const float* __restrict__ bias,
    float* __restrict__ out, int E)
{
    int e = blockIdx.x * blockDim.x + threadIdx.x;
    if (e >= E) return;
    EdgePair p = edges[e];                       // global_load_b128, coalesced
    out[e] = s_tab[p.src] + t_tab[p.dst] + bias[0];
}

// -----------------------------------------------------------------------------
// Launcher
//   d_in[0] = node_embedding  (N*128 f32)
//   d_in[1] = edges           (E*2   i64)
//   d_in[2] = weight          (256   f32)
//   d_in[3] = bias            (1     f32)
//   d_out   = logits          (E     f32)
//   d_ws    = s_tab[N] | t_tab[N]  (2*N*4 = 800 KB)
// -----------------------------------------------------------------------------
extern "C" void kernel_launch(void* const* d_in, const int* in_sizes, int n_in,
                              void* d_out, int out_size, void* d_ws, size_t ws_size,
                              hipStream_t stream)
{
    const float*     emb    = (const float*)d_in[0];
    const EdgePair*  edges  = (const EdgePair*)d_in[1];
    const float*     weight = (const float*)d_in[2];
    const float*     bias   = (const float*)d_in[3];
    float*           out    = (float*)d_out;

    const int N = in_sizes[0] / kD;   // 100000
    const int E = in_sizes[1] / 2;    // 1600000

    float* s_tab = (float*)d_ws;
    float* t_tab = s_tab + N;

    // Phase 1: one wave per 16-node tile, 8 waves (256 threads) per block.
    const int numTiles = (N + 15) / 16;
    const int blocks1  = (numTiles + 7) / 8;
    node_score_wmma<<<blocks1, 256, 0, stream>>>(emb, weight, s_tab, t_tab, N, numTiles);

    // Phase 2: one thread per edge.
    const int threads2 = 256;
    const int blocks2  = (E + threads2 - 1) / threads2;
    edge_logits_kernel<<<blocks2, threads2, 0, stream>>>(edges, s_tab, t_tab, bias, out, E);
}